// DeepGMM_31825707663376
// MI455X (gfx1250) — compile-verified
//
#include <hip/hip_runtime.h>

typedef __attribute__((ext_vector_type(2))) float v2f;
typedef __attribute__((ext_vector_type(8))) float v8f;

#define B_N   4096
#define D_INN 1024
#define D_F   256
#define NCOMP 8
#define NMK   80
#define LOG2PI 1.83787706640934548356f

__device__ __forceinline__ v8f wmma4(v2f a, v2f b, v8f c) {
  // D = A(16x4 f32) * B(4x16 f32) + C(16x16 f32)  -> v_wmma_f32_16x16x4_f32
  return __builtin_amdgcn_wmma_f32_16x16x4_f32(false, a, false, b, (short)0, c,
                                               false, false);
}

// ---------------------------------------------------------------------------
// Kernel 1: feats = relu(x @ W + b)   [4096,1024]x[1024,256] fp32 WMMA GEMM
// Workgroup: 64x64 output tile, 8 waves, 2 WMMA tiles (16x16) per wave.
// LDS: A panel [row][k] stride 66, B panel stored TRANSPOSED [n][k] stride 66
// so every WMMA fragment is one aligned ds_load_b64.
// ---------------------------------------------------------------------------
__global__ void feats_kernel(const float* __restrict__ x,
                             const float* __restrict__ W,
                             const float* __restrict__ bias,
                             float* __restrict__ feats) {
  extern __shared__ float smem[];
  float* As = smem;            // 64 x 66  (x tile, [row][k])
  float* Bs = smem + 64 * 66;  // 64 x 66  (W tile, [n][k]  == B^T)
  const int tid = threadIdx.x;
  const int lane = tid & 31;
  const int w = tid >> 5;
  const int rowbase = blockIdx.y * 64;
  const int colbase = blockIdx.x * 64;

  v8f acc[2] = {};
  const int l15 = lane & 15;
  const int khalf = (lane >> 4) * 2;  // ISA layout: lanes 16-31 hold K=2,3

  for (int kc = 0; kc < D_INN; kc += 64) {
    for (int e = tid; e < 64 * 64; e += 256) {
      int r = e >> 6, c = e & 63;  // consecutive threads -> consecutive c
      As[r * 66 + c] = x[(size_t)(rowbase + r) * D_INN + kc + c];
      // global read coalesced in c (= n), LDS write transposed to [n][k]
      Bs[c * 66 + r] = W[(size_t)(kc + r) * D_F + colbase + c];
    }
    __syncthreads();
    for (int kk = 0; kk < 64; kk += 4) {
      int klo = kk + khalf;
#pragma unroll
      for (int t = 0; t < 2; ++t) {
        int g = w * 2 + t;
        int mt = g >> 2, nt = g & 3;
        v2f a = *(const v2f*)(As + (mt * 16 + l15) * 66 + klo);
        v2f bv = *(const v2f*)(Bs + (nt * 16 + l15) * 66 + klo);
        acc[t] = wmma4(a, bv, acc[t]);
      }
    }
    __syncthreads();
  }

  // epilogue: bias + ReLU, C-layout: VGPR r -> M = r (lanes 0-15) / 8+r (16-31)
  const int hi = lane >> 4;
#pragma unroll
  for (int t = 0; t < 2; ++t) {
    int g = w * 2 + t;
    int mt = g >> 2, nt = g & 3;
    int n = nt * 16 + l15;
    float bb = bias[colbase + n];
#pragma unroll
    for (int r = 0; r < 8; ++r) {
      int mrow = mt * 16 + r + hi * 8;
      float v = acc[t][r] + bb;
      v = v > 0.f ? v : 0.f;
      feats[(size_t)(rowbase + mrow) * D_F + colbase + n] = v;
    }
  }
}

// ---------------------------------------------------------------------------
// Kernel 2: per (m,k): Linv = inv(tril(cov)), logdet = sum(log(diag)).
// One WG of 256 threads per matrix; lane j owns column j; forward
// substitution with the whole Linv held in 263KB of LDS (CDNA5: 320KB/WGP).
// ---------------------------------------------------------------------------
__global__ void trinv_kernel(const float* __restrict__ covs,
                             float* __restrict__ Linv,
                             float* __restrict__ logdet) {
  extern __shared__ float smem[];
  float* Ls = smem;               // [i*257 + j], padded stride
  float* logd = smem + 256 * 257;
  const int j = threadIdx.x;
  const int mk = blockIdx.x;
  const float* Lg = covs + (size_t)mk * (D_F * D_F);

  for (int e = j; e < 256 * 257; e += 256) Ls[e] = 0.f;
  __syncthreads();

  float Ljj = Lg[j * D_F + j];
  Ls[j * 257 + j] = 1.f / Ljj;
  logd[j] = logf(Ljj);
  __syncthreads();

  for (int i = 1; i < D_F; ++i) {
    if (j < i) {
      const float* Lrow = Lg + (size_t)i * D_F;
      float s = 0.f;
      for (int t = j; t < i; ++t) s += Lrow[t] * Ls[t * 257 + j];
      Ls[i * 257 + j] = -s / Lrow[i];
    }
    __syncthreads();
  }

  float* outp = Linv + (size_t)mk * (D_F * D_F);
  for (int i = 0; i < D_F; ++i) outp[(size_t)i * D_F + j] = Ls[i * 257 + j];
  if (j == 0) {
    float s = 0.f;
    for (int t = 0; t < D_F; ++t) s += logd[t];
    logdet[mk] = s;
  }
}

// ---------------------------------------------------------------------------
// Kernel 3: per (m,k, batch-tile of 64):
//   z = Linv * (feats - mu)^T  via fp32 WMMA (M=256, N=64, K=256),
//   quad[n] = sum_rows z^2 (ds_add_f32 LDS atomics),
//   out[b, m, k] = softmax(weights)[m,k] * (-0.5*(quad + D*log(2pi)) - logdet)
// LDS: Linv panel [row][kchunk] stride 66; diff^T stored TRANSPOSED [n][d]
// stride 258 so both WMMA fragments are single aligned ds_load_b64.
// ---------------------------------------------------------------------------
__global__ void gmm_kernel(const float* __restrict__ feats,
                           const float* __restrict__ Linv,
                           const float* __restrict__ logdet,
                           const float* __restrict__ means,
                           const float* __restrict__ weights,
                           float* __restrict__ out) {
  extern __shared__ float smem[];
  float* As = smem;                      // 256 x 66 : Linv[0:256][kc:kc+64]
  float* Bs = smem + 256 * 66;           // 64 x 258 : diff[n][d], full K
  float* quad_s = smem + 256 * 66 + 64 * 258;  // 64
  const int tid = threadIdx.x;
  const int lane = tid & 31;
  const int w = tid >> 5;
  const int mk = blockIdx.y;
  const int b0 = blockIdx.x * 64;
  const float* LinvG = Linv + (size_t)mk * (D_F * D_F);
  const float* mu = means + (size_t)mk * D_F;

  // stage diff = feats - mu for this 64-batch tile (coalesced on d)
  for (int e = tid; e < 64 * 256; e += 256) {
    int n = e >> 8, d = e & 255;
    Bs[n * 258 + d] = feats[(size_t)(b0 + n) * D_F + d] - mu[d];
  }
  if (tid < 64) quad_s[tid] = 0.f;
  __syncthreads();

  v8f acc[8] = {};
  const int l15 = lane & 15;
  const int khalf = (lane >> 4) * 2;

  for (int kc = 0; kc < D_F; kc += 64) {
    for (int e = tid; e < 256 * 64; e += 256) {
      int r = e >> 6, c = e & 63;
      As[r * 66 + c] = LinvG[(size_t)r * D_F + kc + c];
    }
    __syncthreads();
    for (int kk = 0; kk < 64; kk += 4) {
      int klo = kk + khalf;
#pragma unroll
      for (int i = 0; i < 2; ++i) {
        int mt = w * 2 + i;
        v2f a = *(const v2f*)(As + (mt * 16 + l15) * 66 + klo);
#pragma unroll
        for (int nt = 0; nt < 4; ++nt) {
          v2f bv = *(const v2f*)(Bs + (nt * 16 + l15) * 258 + kc + klo);
          acc[i * 4 + nt] = wmma4(a, bv, acc[i * 4 + nt]);
        }
      }
    }
    __syncthreads();
  }

  // colwise sum of z^2: each lane covers 8 consecutive M rows of its tile
#pragma unroll
  for (int t = 0; t < 8; ++t) {
    int nt = t & 3;
    float p = 0.f;
#pragma unroll
    for (int r = 0; r < 8; ++r) p += acc[t][r] * acc[t][r];
    atomicAdd(&quad_s[nt * 16 + l15], p);  // -> ds_add_f32
  }
  __syncthreads();

  if (tid < 64) {
    int m = mk >> 3, kcomp = mk & 7;
    const float* wr = weights + m * NCOMP;
    float wmax = wr[0];
#pragma unroll
    for (int t = 1; t < 8; ++t) wmax = fmaxf(wmax, wr[t]);
    float denom = 0.f;
#pragma unroll
    for (int t = 0; t < 8; ++t) denom += expf(wr[t] - wmax);
    float wmk = expf(wr[kcomp] - wmax) / denom;
    float logp = -0.5f * (quad_s[tid] + 256.0f * LOG2PI) - logdet[mk];
    out[(size_t)(b0 + tid) * NMK + mk] = wmk * logp;
  }
}

// ---------------------------------------------------------------------------
extern "C" void kernel_launch(void* const* d_in, const int* in_sizes, int n_in,
                              void* d_out, int out_size, void* d_ws,
                              size_t ws_size, hipStream_t stream) {
  (void)in_sizes; (void)n_in; (void)out_size; (void)ws_size;
  const float* x       = (const float*)d_in[0];
  const float* W       = (const float*)d_in[1];
  const float* bias    = (const float*)d_in[2];
  const float* means   = (const float*)d_in[3];
  const float* covs    = (const float*)d_in[4];
  const float* weights = (const float*)d_in[5];
  float* out = (float*)d_out;

  float* ws = (float*)d_ws;
  float* feats  = ws;                                   // 4096*256
  float* LinvW  = ws + (size_t)B_N * D_F;               // 80*256*256
  float* logdet = LinvW + (size_t)NMK * D_F * D_F;      // 80

  size_t sh1 = (size_t)(2 * 64 * 66) * sizeof(float);           // 33.8 KB
  feats_kernel<<<dim3(D_F / 64, B_N / 64), 256, sh1, stream>>>(x, W, bias,
                                                               feats);

  size_t sh2 = (size_t)(256 * 257 + 256) * sizeof(float);       // 264 KB LDS
  trinv_kernel<<<dim3(NMK), 256, sh2, stream>>>(covs, LinvW, logdet);

  size_t sh3 = (size_t)(256 * 66 + 64 * 258 + 64) * sizeof(float);  // 134 KB
  gmm_kernel<<<dim3(B_N / 64, NMK), 256, sh3, stream>>>(feats, LinvW, logdet,
                                                        means, weights, out);
}